// Model_47064251629739
// MI455X (gfx1250) — compile-verified
//
#include <hip/hip_runtime.h>
#include <hip/hip_bf16.h>
#include <cstddef>

// ---------------- problem constants (match reference) ----------------
#define NN      50000     // nodes
#define NE      800000    // edges
#define NL      200000    // labeled pairs
#define FEAT    128
#define HID     32
#define HEADS   8
#define H1      (HEADS*HID)   // 256

typedef __attribute__((ext_vector_type(2))) float v2f;
typedef __attribute__((ext_vector_type(8))) float v8f;

// ---------------- helpers ----------------
__device__ __forceinline__ unsigned enc_f32(float f) {
    unsigned u = __float_as_uint(f);
    return (u & 0x80000000u) ? ~u : (u | 0x80000000u);
}
__device__ __forceinline__ float dec_f32(unsigned u) {
    unsigned b = (u & 0x80000000u) ? (u & 0x7FFFFFFFu) : ~u;
    return __uint_as_float(b);
}

// ---------------- zero fill ----------------
__global__ void zero_u32(unsigned* __restrict__ p, int n) {
    int t = blockIdx.x * blockDim.x + threadIdx.x;
    if (t < n) p[t] = 0u;
}

// ---------------- fold attention vectors: V[h*K+k] = sum_c W[k, h*C+c]*att[h,c] ----------------
__global__ void fold_att_kernel(const float* __restrict__ W, const float* __restrict__ att,
                                float* __restrict__ V, int K, int H, int C) {
    int t = blockIdx.x * blockDim.x + threadIdx.x;
    if (t >= K * H) return;
    int h = t / K, k = t - h * K;
    float s = 0.f;
    const float* wr = W + (size_t)k * H * C + h * C;
    const float* ar = att + h * C;
    for (int c = 0; c < C; ++c) s += wr[c] * ar[c];
    V[h * K + k] = s;
}

// ---------------- fp32 WMMA GEMM: C[M,Nc] = A[M,K] @ B[K,Nc] (+bias) ----------------
// one wave per 16x16 tile; K stepped by 4 with V_WMMA_F32_16X16X4_F32
__global__ void wmma_gemm_f32(const float* __restrict__ A, const float* __restrict__ B,
                              const float* __restrict__ bias, float* __restrict__ C,
                              int M, int K, int Nc) {
    const int lane = threadIdx.x;           // 0..31
    const int mt   = blockIdx.x * blockDim.y + threadIdx.y;
    const int row0 = mt * 16;
    const int col0 = blockIdx.y * 16;
    if (row0 >= M) return;                  // wave-uniform (EXEC stays all-ones)
    const int half = lane >> 4;             // 0: K=0,1   1: K=2,3
    const int l16  = lane & 15;

    v8f acc = {};
    const float* Arow = A + (size_t)(row0 + l16) * K + half * 2;
    const float* Bcol = B + (size_t)(half * 2) * Nc + col0 + l16;
    for (int k = 0; k < K; k += 4) {
        v2f a, b;
        a.x = Arow[k + 0];
        a.y = Arow[k + 1];
        b.x = Bcol[(size_t)k * Nc];
        b.y = Bcol[(size_t)(k + 1) * Nc];
        acc = __builtin_amdgcn_wmma_f32_16x16x4_f32(false, a, false, b,
                                                    (short)0, acc, false, false);
    }
    // C/D layout: VGPR r -> M = r (lanes 0-15), M = r+8 (lanes 16-31)
    const int n = col0 + l16;
    const float bv = bias ? bias[n] : 0.0f;
    const int mbase = row0 + half * 8;
#pragma unroll
    for (int r = 0; r < 8; ++r)
        C[(size_t)(mbase + r) * Nc + n] = acc[r] + bv;
}

// ---------------- per-node attention scores: a_s/a_d = H @ V{s,d}^T ----------------
__global__ void scores_kernel(const float* __restrict__ Hrow, const float* __restrict__ Vs,
                              const float* __restrict__ Vd, float* __restrict__ as_,
                              float* __restrict__ ad_, int N, int K, int H) {
    int n = blockIdx.x * blockDim.x + threadIdx.x;
    if (n >= N) return;
    const float* row = Hrow + (size_t)n * K;
    for (int h = 0; h < H; ++h) {
        float s = 0.f, d = 0.f;
        const float* vs = Vs + h * K;
        const float* vd = Vd + h * K;
        for (int k = 0; k < K; ++k) { s += row[k] * vs[k]; d += row[k] * vd[k]; }
        as_[n * H + h] = s;
        ad_[n * H + h] = d;
    }
}

// ---------------- edge pass 1: alpha = leaky_relu(a_s[src]+a_d[dst]); segment max ----------------
__global__ void edge_alpha_max(const int* __restrict__ src, const int* __restrict__ dst,
                               const float* __restrict__ as_, const float* __restrict__ ad_,
                               float* __restrict__ alpha, unsigned* __restrict__ menc,
                               int E, int H) {
    int t = blockIdx.x * blockDim.x + threadIdx.x;
    if (t >= E * H) return;
    int e = t / H, h = t - e * H;
    int s = src[e], d = dst[e];
    float a = as_[s * H + h] + ad_[d * H + h];
    a = (a > 0.f) ? a : 0.2f * a;           // leaky relu, slope 0.2
    alpha[t] = a;
    atomicMax(menc + d * H + h, enc_f32(a));
}

// ---------------- edge pass 2: e = exp(alpha - m[dst]); segment sum ----------------
__global__ void edge_expsum(const int* __restrict__ dst, const unsigned* __restrict__ menc,
                            float* __restrict__ alpha, float* __restrict__ ssum,
                            int E, int H) {
    int t = blockIdx.x * blockDim.x + threadIdx.x;
    if (t >= E * H) return;
    int e = t / H, h = t - e * H;
    int d = dst[e];
    float m = dec_f32(menc[d * H + h]);
    float ex = expf(alpha[t] - m);
    alpha[t] = ex;                           // overwrite with exp value
    atomicAdd(ssum + d * H + h, ex);
}

// ---------------- edge pass 3: out[dst] += xs[src] * coef ----------------
__global__ void edge_scatter(const int* __restrict__ src, const int* __restrict__ dst,
                             const float* __restrict__ xs, const float* __restrict__ alpha,
                             const float* __restrict__ ssum, float* __restrict__ out,
                             int E, int H, int C) {
    const int F = H * C;
    int t = blockIdx.x * blockDim.x + threadIdx.x;
    if (t >= E * F) return;                  // max 204.8M < 2^31
    int e = t / F, c = t - e * F;
    int h = c / C;
    int s = src[e], d = dst[e];
    float coef = alpha[(size_t)e * H + h] / (ssum[d * H + h] + 1e-16f);
    atomicAdd(out + (size_t)d * F + c, xs[(size_t)s * F + c] * coef);
}

// ---------------- bias (+optional relu) in place ----------------
__global__ void bias_act(float* __restrict__ h, const float* __restrict__ bias,
                         int total, int F, int relu) {
    int t = blockIdx.x * blockDim.x + threadIdx.x;
    if (t >= total) return;
    float v = h[t] + bias[t % F];
    h[t] = (relu && v < 0.f) ? 0.f : v;
}

// ---------------- classifier: pred[i] = dot(h[a_i], h[b_i]), 32 ch, wave per pair ----------------
__global__ void classifier(const int* __restrict__ ia, const int* __restrict__ ib,
                           const float* __restrict__ h, float* __restrict__ pred, int L) {
    int wid  = (blockIdx.x * blockDim.x + threadIdx.x) >> 5;
    int lane = threadIdx.x & 31;
    if (wid >= L) return;                    // wave-uniform
    int a = ia[wid], b = ib[wid];
    float p = h[(size_t)a * HID + lane] * h[(size_t)b * HID + lane];
#pragma unroll
    for (int m = 16; m; m >>= 1) p += __shfl_xor(p, m, 32);
    if (lane == 0) pred[wid] = p;
}

// =====================================================================
extern "C" void kernel_launch(void* const* d_in, const int* in_sizes, int n_in,
                              void* d_out, int out_size, void* d_ws, size_t ws_size,
                              hipStream_t stream) {
    const float* x     = (const float*)d_in[0];
    const int*   ei    = (const int*)  d_in[1];   // [2,E]: row0=src, row1=dst
    const int*   eli   = (const int*)  d_in[2];   // [2,L]
    const float* lin_w = (const float*)d_in[3];
    const float* lin_b = (const float*)d_in[4];
    const float* w1s   = (const float*)d_in[5];
    const float* w1d   = (const float*)d_in[6];
    const float* a1s   = (const float*)d_in[7];
    const float* a1d   = (const float*)d_in[8];
    const float* b1    = (const float*)d_in[9];
    const float* w2s   = (const float*)d_in[10];
    const float* w2d   = (const float*)d_in[11];
    const float* a2s   = (const float*)d_in[12];
    const float* a2d   = (const float*)d_in[13];
    const float* b2    = (const float*)d_in[14];
    float* pred = (float*)d_out;

    const int* src = ei;
    const int* dst = ei + NE;

    // ---- workspace layout ----
    float*    h0   = (float*)d_ws;                 // N*32
    float*    xs1  = h0   + (size_t)NN * HID;      // N*256
    float*    as1  = xs1  + (size_t)NN * H1;       // N*8
    float*    ad1  = as1  + (size_t)NN * HEADS;    // N*8
    unsigned* m1   = (unsigned*)(ad1 + (size_t)NN * HEADS);  // N*8
    float*    s1   = (float*)m1 + (size_t)NN * HEADS;        // N*8  (contig after m1)
    float*    al1  = s1   + (size_t)NN * HEADS;    // E*8
    float*    out1 = al1  + (size_t)NE * HEADS;    // N*256 (becomes h1)
    float*    xs2  = out1 + (size_t)NN * H1;       // N*32
    float*    as2  = xs2  + (size_t)NN * HID;      // N
    float*    ad2  = as2  + NN;                    // N
    unsigned* m2   = (unsigned*)(ad2 + NN);        // N
    float*    s2   = (float*)m2 + NN;              // N (contig after m2)
    float*    al2  = s2   + NN;                    // E
    float*    out2 = al2  + NE;                    // N*32 (becomes h2)
    float*    Vs1  = out2 + (size_t)NN * HID;      // 8*32
    float*    Vd1  = Vs1 + HEADS * HID;
    float*    Vs2  = Vd1 + HEADS * HID;            // 256
    float*    Vd2  = Vs2 + H1;

    const int TB = 256;
    auto blks = [](long long n, int tb) { return (unsigned)((n + tb - 1) / tb); };

    // ---- 0. zero accumulators (m1+s1 contiguous; m2+s2 contiguous) ----
    zero_u32<<<blks((long long)NN * HEADS * 2, TB), TB, 0, stream>>>(m1, NN * HEADS * 2);
    zero_u32<<<blks((long long)NN * H1, TB),       TB, 0, stream>>>((unsigned*)out1, NN * H1);
    zero_u32<<<blks(2LL * NN, TB),                 TB, 0, stream>>>(m2, 2 * NN);
    zero_u32<<<blks((long long)NN * HID, TB),      TB, 0, stream>>>((unsigned*)out2, NN * HID);

    // ---- 1. fold attention vectors (eliminates the w*d GEMMs) ----
    fold_att_kernel<<<blks(HID * HEADS, TB), TB, 0, stream>>>(w1s, a1s, Vs1, HID, HEADS, HID);
    fold_att_kernel<<<blks(HID * HEADS, TB), TB, 0, stream>>>(w1d, a1d, Vd1, HID, HEADS, HID);
    fold_att_kernel<<<blks(H1, TB),          TB, 0, stream>>>(w2s, a2s, Vs2, H1, 1, HID);
    fold_att_kernel<<<blks(H1, TB),          TB, 0, stream>>>(w2d, a2d, Vd2, H1, 1, HID);

    // ---- 2. node_lin: h0 = x @ lin_w + lin_b     [50000,128]x[128,32] ----
    {
        dim3 blk(32, 4), grd((NN / 16 + 3) / 4, HID / 16);
        wmma_gemm_f32<<<grd, blk, 0, stream>>>(x, lin_w, lin_b, h0, NN, FEAT, HID);
    }

    // ---- 3. conv1 projection xs1 = h0 @ w1s      [50000,32]x[32,256] ----
    {
        dim3 blk(32, 4), grd((NN / 16 + 3) / 4, H1 / 16);
        wmma_gemm_f32<<<grd, blk, 0, stream>>>(h0, w1s, nullptr, xs1, NN, HID, H1);
    }
    scores_kernel<<<blks(NN, TB), TB, 0, stream>>>(h0, Vs1, Vd1, as1, ad1, NN, HID, HEADS);

    // ---- 4. conv1 edge softmax + aggregate ----
    edge_alpha_max<<<blks((long long)NE * HEADS, TB), TB, 0, stream>>>(src, dst, as1, ad1, al1, m1, NE, HEADS);
    edge_expsum  <<<blks((long long)NE * HEADS, TB), TB, 0, stream>>>(dst, m1, al1, s1, NE, HEADS);
    edge_scatter <<<blks((long long)NE * H1,    TB), TB, 0, stream>>>(src, dst, xs1, al1, s1, out1, NE, HEADS, HID);
    bias_act     <<<blks((long long)NN * H1,    TB), TB, 0, stream>>>(out1, b1, NN * H1, H1, 1);  // h1 = relu(out1+b1)

    // ---- 5. conv2 projection xs2 = h1 @ w2s      [50000,256]x[256,32] ----
    {
        dim3 blk(32, 4), grd((NN / 16 + 3) / 4, HID / 16);
        wmma_gemm_f32<<<grd, blk, 0, stream>>>(out1, w2s, nullptr, xs2, NN, H1, HID);
    }
    scores_kernel<<<blks(NN, TB), TB, 0, stream>>>(out1, Vs2, Vd2, as2, ad2, NN, H1, 1);

    // ---- 6. conv2 edge softmax + aggregate ----
    edge_alpha_max<<<blks((long long)NE, TB), TB, 0, stream>>>(src, dst, as2, ad2, al2, m2, NE, 1);
    edge_expsum  <<<blks((long long)NE, TB), TB, 0, stream>>>(dst, m2, al2, s2, NE, 1);
    edge_scatter <<<blks((long long)NE * HID, TB), TB, 0, stream>>>(src, dst, xs2, al2, s2, out2, NE, 1, HID);
    bias_act     <<<blks((long long)NN * HID, TB), TB, 0, stream>>>(out2, b2, NN * HID, HID, 0);  // h2 = out2+b2

    // ---- 7. classifier: pred = dot(h2[a], h2[b]) ----
    classifier<<<blks((long long)NL * 32, TB), TB, 0, stream>>>(eli, eli + NL, out2, pred, NL);
}